// GNNExpert_18554258719474
// MI455X (gfx1250) — compile-verified
//
#include <hip/hip_runtime.h>
#include <math.h>

// ---------------------------------------------------------------------------
// GAT (2x GATConv + FC + log_softmax) for MI455X (gfx1250, wave32, WMMA).
// Dense GEMMs: guard-free V_WMMA_F32_16X16X4_F32, software-pipelined b64
// fragment loads (double-buffered), weights pre-transposed to column-major,
// FC padded to 48 cols. Edge softmax/aggregation: L2-resident fp32 atomics
// (h + accumulator fit comfortably in the 192MB L2).
// ---------------------------------------------------------------------------

#define GN 50000      // nodes (= 3125 * 16, so M needs no tile guards)
#define GE 800000     // raw edges (self-loops appended logically)
#define GH 8          // heads
#define GD 32         // dim per head
#define GF 256        // H*D
#define GIN 128       // input dim
#define GC 40         // classes
#define GCP 48        // classes padded to 3 WMMA tiles
#define NEG_SLOPE 0.2f

typedef float v2f __attribute__((ext_vector_type(2)));
typedef float v8f __attribute__((ext_vector_type(8)));

// ---------------- weight transpose (+ zero pad) ----------------------------
// WT[n*K + k] = (n < Nc) ? W[k*Nc + n] : 0   for n in [0, NcPad)
__global__ void gat_transpose_pad(const float* __restrict__ W,
                                  float* __restrict__ WT,
                                  int K, int Nc, int NcPad) {
    int i = blockIdx.x * blockDim.x + threadIdx.x;   // n*K + k
    if (i >= NcPad * K) return;
    int n = i / K;
    int k = i - n * K;
    WT[i] = (n < Nc) ? W[(size_t)k * Nc + n] : 0.0f;
}

// ---------------- fp32 WMMA GEMM: C[M,NcPad] = A[M,K] @ B -------------------
// BT is column-major B: BT[n*K + k] = B[k][n]. One wave32 computes a
// 16 x (16*NT) strip. K is compile-time -> fully unrolled, double-buffered
// fragments so loads for k-step i+1 overlap the WMMAs of k-step i.
// Requires: M % 16 == 0, K % 4 == 0, NcPad % (16*NT) == 0 (no guards).
template<int NT, int K>
__global__ void gat_gemm_wmma(const float* __restrict__ A,
                              const float* __restrict__ BT,
                              float* __restrict__ C,
                              int M, int NcPad) {
    int wave = (int)((blockIdx.x * blockDim.x + threadIdx.x) >> 5);
    int lane = (int)(threadIdx.x & 31);
    int tiles_m = M >> 4;
    int ngroups = NcPad / (16 * NT);
    if (wave >= tiles_m * ngroups) return;     // wave-uniform: EXEC stays full
    int tm = wave / ngroups;
    int ng = wave - tm * ngroups;
    int m0 = tm << 4;
    int n0 = ng * (16 * NT);

    int half = lane >> 4;                      // K-slice owner (ISA 16x4 A layout)
    int r    = lane & 15;

    const float* ap = A + (size_t)(m0 + r) * K + (half << 1);
    const float* bp[NT];
#pragma unroll
    for (int t = 0; t < NT; ++t)
        bp[t] = BT + (size_t)(n0 + t * 16 + r) * K + (half << 1);

    v8f acc[NT];
#pragma unroll
    for (int t = 0; t < NT; ++t) acc[t] = (v8f){};

    // prologue: fragments for k0 = 0
    v2f a0 = *(const v2f*)(ap);
    v2f b0[NT];
#pragma unroll
    for (int t = 0; t < NT; ++t) b0[t] = *(const v2f*)(bp[t]);

#pragma unroll
    for (int k0 = 0; k0 < K - 4; k0 += 4) {
        // prefetch k-step k0+4 while the WMMAs below consume k0's fragments
        v2f a1 = *(const v2f*)(ap + k0 + 4);
        v2f b1[NT];
#pragma unroll
        for (int t = 0; t < NT; ++t) b1[t] = *(const v2f*)(bp[t] + k0 + 4);
#pragma unroll
        for (int t = 0; t < NT; ++t)
            acc[t] = __builtin_amdgcn_wmma_f32_16x16x4_f32(
                false, a0, false, b0[t], (short)0, acc[t], false, false);
        a0 = a1;
#pragma unroll
        for (int t = 0; t < NT; ++t) b0[t] = b1[t];
    }
    // epilogue: last k-step
#pragma unroll
    for (int t = 0; t < NT; ++t)
        acc[t] = __builtin_amdgcn_wmma_f32_16x16x4_f32(
            false, a0, false, b0[t], (short)0, acc[t], false, false);

    // C/D layout: VGPR j -> row m0 + j + 8*half, col n0 + 16*t + r
#pragma unroll
    for (int t = 0; t < NT; ++t) {
        int bcol = n0 + t * 16 + r;
#pragma unroll
        for (int j = 0; j < 8; ++j) {
            int row = m0 + j + (half << 3);
            C[(size_t)row * NcPad + bcol] = acc[t][j];
        }
    }
}

// ---------------- helpers: monotone uint encoding for float atomicMax ------
__device__ __forceinline__ unsigned fkey(float f) {
    unsigned u = __float_as_uint(f);
    return (u & 0x80000000u) ? ~u : (u | 0x80000000u);
}
__device__ __forceinline__ float fdec(unsigned k) {
    return (k & 0x80000000u) ? __uint_as_float(k ^ 0x80000000u)
                             : __uint_as_float(~k);
}
__device__ __forceinline__ float lrelu(float x) {
    return x > 0.0f ? x : NEG_SLOPE * x;
}

// per-(node,head): e_src/e_dst dots, init max-keys & sums, zero accumulator
__global__ void gat_prep(const float* __restrict__ h,
                         const float* __restrict__ a_src,
                         const float* __restrict__ a_dst,
                         float* __restrict__ e_src,
                         float* __restrict__ e_dst,
                         unsigned* __restrict__ mkey,
                         float* __restrict__ ssum,
                         float* __restrict__ outbuf) {
    int idx = blockIdx.x * blockDim.x + threadIdx.x;   // n*H + head
    if (idx >= GN * GH) return;
    int n  = idx >> 3;
    int hh = idx & 7;
    const float* hp = h + (size_t)n * GF + hh * GD;
    const float* as = a_src + hh * GD;
    const float* ad = a_dst + hh * GD;
    float* op = outbuf + (size_t)n * GF + hh * GD;
    float s0 = 0.0f, s1 = 0.0f;
#pragma unroll 8
    for (int d = 0; d < GD; ++d) {
        float v = hp[d];
        s0 += v * as[d];
        s1 += v * ad[d];
        op[d] = 0.0f;
    }
    e_src[idx] = s0;
    e_dst[idx] = s1;
    mkey[idx]  = 0u;      // below every real float's key
    ssum[idx]  = 0.0f;
}

__device__ __forceinline__ void edge_endpoints(const int* __restrict__ ei,
                                               int e, int* s, int* d) {
    if (e < GE) { *s = ei[e]; *d = ei[GE + e]; }
    else        { *s = *d = e - GE; }            // appended self-loops
}

// pass 1: segment max of leaky-relu scores (ordered-uint atomicMax)
__global__ void gat_edge_max(const int* __restrict__ ei,
                             const float* __restrict__ e_src,
                             const float* __restrict__ e_dst,
                             unsigned* __restrict__ mkey) {
    int e = blockIdx.x * blockDim.x + threadIdx.x;
    if (e >= GE + GN) return;
    int s, d;
    edge_endpoints(ei, e, &s, &d);
#pragma unroll
    for (int hh = 0; hh < GH; ++hh) {
        float v = lrelu(e_src[s * GH + hh] + e_dst[d * GH + hh]);
        atomicMax(&mkey[d * GH + hh], fkey(v));
    }
}

// pass 2: p = exp(e - m[dst]); segment sum
__global__ void gat_edge_exp(const int* __restrict__ ei,
                             const float* __restrict__ e_src,
                             const float* __restrict__ e_dst,
                             const unsigned* __restrict__ mkey,
                             float* __restrict__ p,
                             float* __restrict__ ssum) {
    int e = blockIdx.x * blockDim.x + threadIdx.x;
    if (e >= GE + GN) return;
    int s, d;
    edge_endpoints(ei, e, &s, &d);
#pragma unroll
    for (int hh = 0; hh < GH; ++hh) {
        float v  = lrelu(e_src[s * GH + hh] + e_dst[d * GH + hh]);
        float mm = fdec(mkey[d * GH + hh]);
        float pv = expf(v - mm);
        p[(size_t)e * GH + hh] = pv;
        atomicAdd(&ssum[d * GH + hh], pv);
    }
}

// pass 3: out[dst,h,:] += h[src,h,:] * alpha  (one 256-thread block per edge)
__global__ void gat_edge_agg(const int* __restrict__ ei,
                             const float* __restrict__ h,
                             const float* __restrict__ p,
                             const float* __restrict__ ssum,
                             float* __restrict__ outbuf) {
    int e   = blockIdx.x;
    int tid = threadIdx.x;       // 256 = GH * GD
    int hh  = tid >> 5;
    int dd  = tid & 31;
    int s, d;
    edge_endpoints(ei, e, &s, &d);
    float alpha = p[(size_t)e * GH + hh] / (ssum[d * GH + hh] + 1e-16f);
    float v = h[(size_t)s * GF + hh * GD + dd] * alpha;
    atomicAdd(&outbuf[(size_t)d * GF + hh * GD + dd], v);
}

// out = relu(out + bias)
__global__ void gat_bias_relu(float* __restrict__ buf,
                              const float* __restrict__ b) {
    int i = blockIdx.x * blockDim.x + threadIdx.x;
    if (i >= GN * GF) return;
    float v = buf[i] + b[i & (GF - 1)];
    buf[i] = v > 0.0f ? v : 0.0f;
}

// per-row log_softmax over GC classes (logits stored with stride GCP)
__global__ void gat_logsoftmax(const float* __restrict__ logits,
                               const float* __restrict__ fcb,
                               float* __restrict__ out) {
    int n = blockIdx.x * blockDim.x + threadIdx.x;
    if (n >= GN) return;
    const float* row = logits + (size_t)n * GCP;
    float mx = -3.4e38f;
    for (int c = 0; c < GC; ++c) {
        float t = row[c] + fcb[c];
        mx = t > mx ? t : mx;
    }
    float sum = 0.0f;
    for (int c = 0; c < GC; ++c) sum += expf(row[c] + fcb[c] - mx);
    float l = logf(sum) + mx;
    float* orow = out + (size_t)n * GC;
    for (int c = 0; c < GC; ++c) orow[c] = (row[c] + fcb[c]) - l;
}

// ---------------------------------------------------------------------------
extern "C" void kernel_launch(void* const* d_in, const int* in_sizes, int n_in,
                              void* d_out, int out_size, void* d_ws, size_t ws_size,
                              hipStream_t stream) {
    const float* x   = (const float*)d_in[0];
    const int*   ei  = (const int*)d_in[1];     // [2, E] row-major: src then dst
    const float* W1  = (const float*)d_in[2];
    const float* a1s = (const float*)d_in[3];
    const float* a1d = (const float*)d_in[4];
    const float* b1  = (const float*)d_in[5];
    const float* W2  = (const float*)d_in[6];
    const float* a2s = (const float*)d_in[7];
    const float* a2d = (const float*)d_in[8];
    const float* b2  = (const float*)d_in[9];
    const float* fcW = (const float*)d_in[10];
    const float* fcb = (const float*)d_in[11];
    float* out = (float*)d_out;

    // workspace carve-up (fully rewritten every call; no cross-call state)
    char* ws = (char*)d_ws;
    size_t off = 0;
    auto carve = [&](size_t bytes) -> void* {
        void* ptr = (void*)(ws + off);
        off += (bytes + 255) & ~(size_t)255;
        return ptr;
    };
    float*    bufA  = (float*)   carve((size_t)GN * GF * sizeof(float)); // h / logits
    float*    bufB  = (float*)   carve((size_t)GN * GF * sizeof(float)); // aggregated
    float*    e_src = (float*)   carve((size_t)GN * GH * sizeof(float));
    float*    e_dst = (float*)   carve((size_t)GN * GH * sizeof(float));
    unsigned* mkey  = (unsigned*)carve((size_t)GN * GH * sizeof(unsigned));
    float*    ssum  = (float*)   carve((size_t)GN * GH * sizeof(float));
    float*    pbuf  = (float*)   carve((size_t)(GE + GN) * GH * sizeof(float));
    float*    W1T   = (float*)   carve((size_t)GF  * GIN * sizeof(float)); // [256,128]
    float*    W2T   = (float*)   carve((size_t)GF  * GF  * sizeof(float)); // [256,256]
    float*    fcWT  = (float*)   carve((size_t)GCP * GF  * sizeof(float)); // [48,256]

    const int Etot = GE + GN;
    dim3 blk(256);
    int eblocks  = (Etot + 255) / 256;
    int nhblocks = (GN * GH + 255) / 256;
    int nfblocks = (GN * GF + 255) / 256;

    // pre-transpose weights to column-major (pad FC to 48 cols)
    gat_transpose_pad<<<(GF * GIN + 255) / 256, blk, 0, stream>>>(W1, W1T, GIN, GF, GF);
    gat_transpose_pad<<<(GF * GF + 255) / 256, blk, 0, stream>>>(W2, W2T, GF, GF, GF);
    gat_transpose_pad<<<(GCP * GF + 255) / 256, blk, 0, stream>>>(fcW, fcWT, GF, GC, GCP);

    auto edge_phase = [&](const float* hbuf, const float* as, const float* ad,
                          const float* b, float* obuf) {
        gat_prep<<<nhblocks, blk, 0, stream>>>(hbuf, as, ad, e_src, e_dst,
                                               mkey, ssum, obuf);
        gat_edge_max<<<eblocks, blk, 0, stream>>>(ei, e_src, e_dst, mkey);
        gat_edge_exp<<<eblocks, blk, 0, stream>>>(ei, e_src, e_dst, mkey, pbuf, ssum);
        gat_edge_agg<<<Etot, blk, 0, stream>>>(ei, hbuf, pbuf, ssum, obuf);
        gat_bias_relu<<<nfblocks, blk, 0, stream>>>(obuf, b);
    };

    int waves256 = (GN >> 4) * (GF / 64);            // NT=4 -> 64-wide strips
    int blocks256 = (waves256 + 7) / 8;

    // Layer 1: x[N,128] -> bufB[N,256]
    gat_gemm_wmma<4, GIN><<<blocks256, 256, 0, stream>>>(x, W1T, bufA, GN, GF);
    edge_phase(bufA, a1s, a1d, b1, bufB);
    // Layer 2: bufB -> bufB (hbuf=bufA; stream order makes in-place obuf safe)
    gat_gemm_wmma<4, GF><<<blocks256, 256, 0, stream>>>(bufB, W2T, bufA, GN, GF);
    edge_phase(bufA, a2s, a2d, b2, bufB);
    // FC logits (padded width 48) into bufA, then log_softmax to d_out
    {
        int waves = (GN >> 4);                        // NT=3 -> one 48-wide strip
        gat_gemm_wmma<3, GF><<<(waves + 7) / 8, 256, 0, stream>>>(bufB, fcWT, bufA, GN, GCP);
    }
    gat_logsoftmax<<<(GN + 255) / 256, blk, 0, stream>>>(bufA, fcb, out);
}